// RNN_64819646431716
// MI455X (gfx1250) — compile-verified
//
#include <hip/hip_runtime.h>
#include <hip/hip_bf16.h>

// RNN scan on MI455X (gfx1250, wave32, WMMA + Tensor Data Mover).
// Phase 1: xs = einsum('bti,hi->tbh', x, w_x) + b_i2h  (parallel, 137 GFLOP)
// Phase 2: h_t = xs[t] + h_{t-1} @ w_h^T               (1024 sequential GEMMs)
// Phase 3: log_softmax(h_T @ w_h2o^T + b_h2o)
//
// Matmuls in f32 via V_WMMA_F32_16X16X4_F32 (reference is f32; 1024 recurrent
// steps make operand rounding risky). The shared A tile of each workgroup is
// staged into LDS once via TENSOR_LOAD_TO_LDS (TDM), with TDM padding
// (interval 256 dw, amount 2 dw -> row stride 516 dw, 516%64=4) so the 16
// lanes of an A fragment hit 16 distinct LDS banks.
//
// Workspace: xs (T*B*H f32 = 512MB) + two B*H ping-pong h buffers.

typedef float v2f __attribute__((ext_vector_type(2)));
typedef float v8f __attribute__((ext_vector_type(8)));
typedef unsigned int u32x4 __attribute__((ext_vector_type(4)));
typedef unsigned int u32x8 __attribute__((ext_vector_type(8)));

constexpr int Bn = 128;
constexpr int Tn = 1024;
constexpr int In = 512;
constexpr int Hn = 1024;
constexpr int On = 256;
constexpr int WROW = In + Hn;   // row stride of w_i2h (H, I+H)
constexpr int LDSROW = 516;     // 512 data dw + 2 pads * 2 dw (TDM padding)

// ---------------------------------------------------------------------------
// V_WMMA_F32_16X16X4_F32 fragments (ISA 7.12.2):
//   A 16x4 (MxK): lane l holds row m=l&15, K = 2*(l>>4)+{0,1}  -> float2
//   B 4x16 (KxN): lane l holds col n=l&15, K = 2*(l>>4)+{0,1}; with B = W^T
//                 this is W[n][k..k+1], contiguous -> float2
//   C/D 16x16:    reg j: row m = j + 8*(l>>4), col n = l&15
// ---------------------------------------------------------------------------
__device__ __forceinline__ v8f wmma_f32(v2f a, v2f b, v8f c) {
  return __builtin_amdgcn_wmma_f32_16x16x4_f32(
      false, a, false, b, (short)0, c, false, false);
}

// ---------------------------------------------------------------------------
// TDM: DMA a rows x cols f32 tile (row stride in elements) from global into
// LDS at lds_byte_off, with padding: 2 DWORDs inserted every 256 DWORDs.
// D# group0 (4 SGPRs) + group1 (8 SGPRs); 2D tile -> VADDR2/3 omitted (NULL).
// Issue from ONE wave; caller waits TENSORcnt then barriers.
// ---------------------------------------------------------------------------
__device__ __forceinline__ void tdm_load_tile_f32(
    unsigned lds_byte_off, const float* gptr, unsigned rows, unsigned cols,
    unsigned long long row_stride_elems) {
  const unsigned long long ga = (unsigned long long)(uintptr_t)gptr;
  u32x4 g0;
  g0.x = 1u;                                  // count=1 (valid), load, user
  g0.y = lds_byte_off;                        // D#.lds_addr (bytes)
  g0.z = (unsigned)ga;                        // global_addr[31:0]
  g0.w = ((unsigned)(ga >> 32) & 0x01ffffffu) // global_addr[56:32]
         | (2u << 30);                        // type = 2 ("image")
  u32x8 g1;
  g1[0] = (2u << 16)                          // data_size = 4 bytes
        | (1u << 20)                          // pad_enable
        | (7u << 22)                          // pad_interval = 256 DWORDs
        | (1u << 25);                         // pad_amount   = 2 DWORDs
  g1[1] = (cols & 0xffffu) << 16;             // tensor_dim0 [79:48] lo16
  g1[2] = (cols >> 16) | ((rows & 0xffffu) << 16);  // dim0 hi16 | dim1 lo16
  g1[3] = (rows >> 16) | ((cols & 0xffffu) << 16);  // dim1 hi16 | tile_dim0
  g1[4] = rows & 0xffffu;                     // tile_dim1 (tile_dim2 = 0)
  g1[5] = (unsigned)row_stride_elems;         // tensor_dim0_stride[31:0]
  g1[6] = (unsigned)(row_stride_elems >> 32) & 0xffffu;  // stride[47:32]
  g1[7] = 0u;                                 // tensor_dim1_stride unused (2D)
  asm volatile("tensor_load_to_lds %0, %1" ::"s"(g0), "s"(g1) : "memory");
}

// 512-wide K slab: A fragments from LDS (ds_load_b64), B streamed from
// global/L2, 4 independent accumulators (breaks the WMMA RAW chain).
// alds points at lds_row(m) + kb; pad offset +2 applies for k >= 256.
__device__ __forceinline__ void gemm_strip_512(
    const float* __restrict__ alds, const float* __restrict__ br0,
    const float* __restrict__ br1, const float* __restrict__ br2,
    const float* __restrict__ br3, v8f& c0, v8f& c1, v8f& c2, v8f& c3) {
#pragma unroll 4
  for (int k = 0; k < 256; k += 4) {
    v2f a = *(const v2f*)(alds + k);
    c0 = wmma_f32(a, *(const v2f*)(br0 + k), c0);
    c1 = wmma_f32(a, *(const v2f*)(br1 + k), c1);
    c2 = wmma_f32(a, *(const v2f*)(br2 + k), c2);
    c3 = wmma_f32(a, *(const v2f*)(br3 + k), c3);
  }
#pragma unroll 4
  for (int k = 256; k < 512; k += 4) {
    v2f a = *(const v2f*)(alds + k + 2);      // skip mid-row TDM pad
    c0 = wmma_f32(a, *(const v2f*)(br0 + k), c0);
    c1 = wmma_f32(a, *(const v2f*)(br1 + k), c1);
    c2 = wmma_f32(a, *(const v2f*)(br2 + k), c2);
    c3 = wmma_f32(a, *(const v2f*)(br3 + k), c3);
  }
}

// ---------------------------------------------------------------------------
// Phase 1: xs[t][b][h] = sum_i x[b][t][i]*w_i2h[h][i] + b_i2h[h]
// Block = 8 waves sharing one 16-row A tile (TDM -> LDS once); wave w owns a
// 64-col h strip. Grid: t(1024) x mt(8) x half(2) = 16384 blocks.
// ---------------------------------------------------------------------------
__global__ __launch_bounds__(256) void xw_kernel(
    const float* __restrict__ x, const float* __restrict__ w_i2h,
    const float* __restrict__ b_i2h, float* __restrict__ xs) {
  __shared__ float lds_a[16 * LDSROW];
  const int tid = threadIdx.x;
  const int lane = tid & 31;
  const int widx = tid >> 5;
  const int bid = blockIdx.x;
  const int t = bid >> 4;
  const int mt = (bid >> 1) & 7;
  const int half = bid & 1;
  const int b0 = mt * 16;
  const int h0 = (half * 8 + widx) * 64;
  const int m = lane & 15;
  const int kb = (lane >> 4) * 2;
  const int hi8 = (lane >> 4) * 8;

  if (tid < 32) {
    tdm_load_tile_f32((unsigned)(uintptr_t)lds_a,
                      x + (size_t)b0 * (Tn * In) + (size_t)t * In,
                      16, 512, (unsigned long long)Tn * In);
    __builtin_amdgcn_s_wait_tensorcnt(0);
  }
  __syncthreads();

  const float* alds = lds_a + m * LDSROW + kb;
  const float* br0 = w_i2h + (size_t)(h0 + 0 * 16 + m) * WROW + kb;
  const float* br1 = w_i2h + (size_t)(h0 + 1 * 16 + m) * WROW + kb;
  const float* br2 = w_i2h + (size_t)(h0 + 2 * 16 + m) * WROW + kb;
  const float* br3 = w_i2h + (size_t)(h0 + 3 * 16 + m) * WROW + kb;

  v8f c0 = {}, c1 = {}, c2 = {}, c3 = {};
  gemm_strip_512(alds, br0, br1, br2, br3, c0, c1, c2, c3);

  const float bias0 = b_i2h[h0 + 0 * 16 + m];
  const float bias1 = b_i2h[h0 + 1 * 16 + m];
  const float bias2 = b_i2h[h0 + 2 * 16 + m];
  const float bias3 = b_i2h[h0 + 3 * 16 + m];
  float* orow = xs + ((size_t)t * Bn + (b0 + hi8)) * Hn + h0 + m;
#pragma unroll
  for (int j = 0; j < 8; ++j) {
    orow[(size_t)j * Hn + 0 * 16] = c0[j] + bias0;
    orow[(size_t)j * Hn + 1 * 16] = c1[j] + bias1;
    orow[(size_t)j * Hn + 2 * 16] = c2[j] + bias2;
    orow[(size_t)j * Hn + 3 * 16] = c3[j] + bias3;
  }
}

// ---------------------------------------------------------------------------
// Phase 2 (one timestep): hout = xs_t + hin @ w_h^T,  w_h = w_i2h[:, I:]
// Block = 8 waves sharing one 16-row h tile; K=1024 in two 512 chunks, each
// TDM-staged into the same 33KB LDS slab. Grid: mt(8) x half(2) = 16 blocks.
// w_h (4MB) and h (512KB) stay L2-resident across all 1023 step launches.
// ---------------------------------------------------------------------------
__global__ __launch_bounds__(256) void step_kernel(
    const float* __restrict__ xs_t, const float* __restrict__ hin,
    float* __restrict__ hout, const float* __restrict__ w_i2h) {
  __shared__ float lds_a[16 * LDSROW];
  const int tid = threadIdx.x;
  const int lane = tid & 31;
  const int widx = tid >> 5;
  const int bid = blockIdx.x;
  const int mt = bid >> 1;
  const int half = bid & 1;
  const int b0 = mt * 16;
  const int h0 = (half * 8 + widx) * 64;
  const int m = lane & 15;
  const int kb = (lane >> 4) * 2;
  const int hi8 = (lane >> 4) * 8;

  // Seed accumulators with the xs[t] tile (fused add).
  v8f c0, c1, c2, c3;
  const float* srow = xs_t + (size_t)(b0 + hi8) * Hn + h0 + m;
#pragma unroll
  for (int j = 0; j < 8; ++j) {
    c0[j] = srow[(size_t)j * Hn + 0 * 16];
    c1[j] = srow[(size_t)j * Hn + 1 * 16];
    c2[j] = srow[(size_t)j * Hn + 2 * 16];
    c3[j] = srow[(size_t)j * Hn + 3 * 16];
  }

  for (int c = 0; c < 2; ++c) {
    __syncthreads();  // previous chunk fully consumed before overwrite
    if (tid < 32) {
      tdm_load_tile_f32((unsigned)(uintptr_t)lds_a,
                        hin + (size_t)b0 * Hn + c * 512,
                        16, 512, (unsigned long long)Hn);
      __builtin_amdgcn_s_wait_tensorcnt(0);
    }
    __syncthreads();  // tile visible to all 8 waves

    const float* alds = lds_a + m * LDSROW + kb;
    const int ko = In + c * 512 + kb;  // w_h column offset within w_i2h row
    const float* br0 = w_i2h + (size_t)(h0 + 0 * 16 + m) * WROW + ko;
    const float* br1 = w_i2h + (size_t)(h0 + 1 * 16 + m) * WROW + ko;
    const float* br2 = w_i2h + (size_t)(h0 + 2 * 16 + m) * WROW + ko;
    const float* br3 = w_i2h + (size_t)(h0 + 3 * 16 + m) * WROW + ko;
    gemm_strip_512(alds, br0, br1, br2, br3, c0, c1, c2, c3);
  }

  float* orow = hout + (size_t)(b0 + hi8) * Hn + h0 + m;
#pragma unroll
  for (int j = 0; j < 8; ++j) {
    orow[(size_t)j * Hn + 0 * 16] = c0[j];
    orow[(size_t)j * Hn + 1 * 16] = c1[j];
    orow[(size_t)j * Hn + 2 * 16] = c2[j];
    orow[(size_t)j * Hn + 3 * 16] = c3[j];
  }
}

// ---------------------------------------------------------------------------
// Phase 3: out[b][o] = log_softmax(h_T[b] . w_h2o[o] + b_h2o[o]).
// One block per batch row; 256 threads = 256 logits; LDS tree reductions.
// ---------------------------------------------------------------------------
__global__ __launch_bounds__(256) void out_kernel(
    const float* __restrict__ hfin, const float* __restrict__ w_h2o,
    const float* __restrict__ b_h2o, float* __restrict__ out) {
  const int b = blockIdx.x;
  const int o = threadIdx.x;

  const float4* hr = (const float4*)(hfin + (size_t)b * Hn);
  const float4* wr = (const float4*)(w_h2o + (size_t)o * Hn);
  float acc = b_h2o[o];
#pragma unroll 4
  for (int k = 0; k < Hn / 4; ++k) {
    float4 hv = hr[k];
    float4 wv = wr[k];
    acc += hv.x * wv.x + hv.y * wv.y + hv.z * wv.z + hv.w * wv.w;
  }

  __shared__ float red[On];
  red[o] = acc;
  __syncthreads();
#pragma unroll
  for (int s = On / 2; s > 0; s >>= 1) {
    if (o < s) red[o] = fmaxf(red[o], red[o + s]);
    __syncthreads();
  }
  const float mx = red[0];
  __syncthreads();
  red[o] = expf(acc - mx);
  __syncthreads();
#pragma unroll
  for (int s = On / 2; s > 0; s >>= 1) {
    if (o < s) red[o] += red[o + s];
    __syncthreads();
  }
  const float lse = mx + logf(red[0]);
  out[(size_t)b * On + o] = acc - lse;
}

// ---------------------------------------------------------------------------

extern "C" void kernel_launch(void* const* d_in, const int* in_sizes, int n_in,
                              void* d_out, int out_size, void* d_ws,
                              size_t ws_size, hipStream_t stream) {
  const float* x = (const float*)d_in[0];      // (B,T,I)
  const float* w_i2h = (const float*)d_in[1];  // (H, I+H)
  const float* b_i2h = (const float*)d_in[2];  // (H)
  const float* w_h2o = (const float*)d_in[3];  // (O, H)
  const float* b_h2o = (const float*)d_in[4];  // (O)
  float* out = (float*)d_out;                  // (B,O)

  float* xs = (float*)d_ws;                // T*B*H f32 (512MB)
  float* hb0 = xs + (size_t)Tn * Bn * Hn;  // B*H ping
  float* hb1 = hb0 + (size_t)Bn * Hn;      // B*H pong

  // Phase 1: 16384 blocks x 8 waves.
  xw_kernel<<<Tn * 16, 256, 0, stream>>>(x, w_i2h, b_i2h, xs);

  // Phase 2: h0 = 0 => h1 = xs[0]; scan starts at t=1 with hin = xs[0]
  // (contiguous (B,H) block at xs + 0).
  for (int t = 1; t < Tn; ++t) {
    const float* hin = (t == 1) ? xs : ((t & 1) ? hb0 : hb1);
    float* hout = (t & 1) ? hb1 : hb0;
    step_kernel<<<16, 256, 0, stream>>>(xs + (size_t)t * Bn * Hn, hin, hout,
                                        w_i2h);
  }

  // T-1 = 1023 is odd -> final hidden state lives in hb1.
  out_kernel<<<Bn, On, 0, stream>>>(hb1, w_h2o, b_h2o, out);
}